// LargeEmbedding_90494960927132
// MI455X (gfx1250) — compile-verified
//
#include <hip/hip_runtime.h>
#include <hip/hip_bf16.h>

// LargeEmbedding gather for MI455X (gfx1250, wave32).
//
// out[slot, 0:64] = pages_flat[ idx[slot]*64 : idx[slot]*64 + 64 ]
// (page_id*PAGE_SIZE + row_id == idx, so the two-level gather is a flat gather)
//
// Pure HBM-bandwidth problem (~68 MB moved, ~3 us floor at 23.3 TB/s).
// Design:
//   * 16 lanes per row, float4 (b128) per lane -> every vmem instruction moves
//     whole 256B row segments (full cacheline utilization, read and write).
//   * UNROLL=4 rows per 16-lane group: 4 outstanding global_load_b128 per
//     half-wave (~2KB in flight per wave) to cover the ~14MB latency-BW product.
//   * Non-temporal hints everywhere: the 256MB table exceeds the 192MB L2 and
//     indices are near-unique (no reuse); output is write-once streaming.
//   * The 4 consecutive int32 indices per group are fetched as one b128 load.
//   * Indices treated as unsigned: zero-extend folds into v_lshl_add_u64,
//     eliminating per-index sign-extension VALU ops seen in the prior build.

typedef __attribute__((ext_vector_type(4))) float    v4f;
typedef __attribute__((ext_vector_type(4))) unsigned v4u;

static constexpr int LANES_PER_ROW = 16;  // 16 lanes * float4 = 64 floats/row
static constexpr int UNROLL        = 4;   // rows per 16-lane group
static constexpr int BLOCK         = 256; // 8 wave32 per block

__global__ __launch_bounds__(BLOCK)
void large_embedding_gather(const unsigned* __restrict__ indices,
                            const v4f* __restrict__ table,  // [n_rows*16] float4
                            v4f* __restrict__ out,          // [n_slots*16] float4
                            int n_slots)
{
    const int gtid  = blockIdx.x * BLOCK + threadIdx.x;
    const int group = gtid >> 4;        // one 16-lane group per UNROLL rows
    const int lane  = gtid & 15;
    const int base  = group * UNROLL;

    if (base >= n_slots) return;

    if (base + UNROLL <= n_slots) {
        // Fast path: one b128 load grabs 4 consecutive int32 indices
        // (base is a multiple of 4 -> 16B aligned). Read-once -> NT.
        const v4u rows = __builtin_nontemporal_load(
            reinterpret_cast<const v4u*>(indices + base));

        v4f v[UNROLL];
        // Issue all gathers before any store: 4 outstanding b128 loads
        // per half-wave for memory-level parallelism.
        #pragma unroll
        for (int u = 0; u < UNROLL; ++u) {
            const size_t src = (size_t)rows[u] * LANES_PER_ROW + lane;
            v[u] = __builtin_nontemporal_load(&table[src]);
        }
        #pragma unroll
        for (int u = 0; u < UNROLL; ++u) {
            const size_t dst = (size_t)(base + u) * LANES_PER_ROW + lane;
            __builtin_nontemporal_store(v[u], &out[dst]);
        }
    } else {
        // Tail (not taken for 131072 slots, kept for generality).
        for (int s = base; s < n_slots; ++s) {
            const unsigned row = indices[s];
            const v4f v = __builtin_nontemporal_load(
                &table[(size_t)row * LANES_PER_ROW + lane]);
            __builtin_nontemporal_store(
                v, &out[(size_t)s * LANES_PER_ROW + lane]);
        }
    }
}

extern "C" void kernel_launch(void* const* d_in, const int* in_sizes, int n_in,
                              void* d_out, int out_size, void* d_ws, size_t ws_size,
                              hipStream_t stream)
{
    (void)n_in; (void)d_ws; (void)ws_size; (void)out_size;

    const unsigned* indices = (const unsigned*)d_in[0];    // [16*8192] int32 (values in [0, 1e6))
    const float*    pages   = (const float*)d_in[1];       // [8*125000*64] f32 (flat table)
    float*          out     = (float*)d_out;               // [16*8192*64] f32

    const int n_slots = in_sizes[0];                       // 131072

    const int groups        = (n_slots + UNROLL - 1) / UNROLL;
    const long long threads = (long long)groups * LANES_PER_ROW;
    const int blocks        = (int)((threads + BLOCK - 1) / BLOCK);

    large_embedding_gather<<<blocks, BLOCK, 0, stream>>>(
        indices,
        reinterpret_cast<const v4f*>(pages),
        reinterpret_cast<v4f*>(out),
        n_slots);
}